// HeightmapNormalsLoss_56934086476418
// MI455X (gfx1250) — compile-verified
//
#include <hip/hip_runtime.h>
#include <stdint.h>

// Heightmap Sobel-normals L1 loss for MI455X (gfx1250).
// HBM-bound stencil (~67 MB traffic -> ~3us roofline at 23.3 TB/s).
// CDNA5 path: async global->LDS b128 staging (ASYNCcnt), separable Sobel
// out of LDS, hardware v_sqrt/v_rsq transcendentals (TRANS32, co-executing),
// deterministic two-stage reduction (graph-replay bit-stable).

#define IMG_HW     512
#define TW         256          // tile width  (== blockDim.x, one column/thread)
#define TH         16           // tile height
#define HALO_H     (TH + 2)     // 18 rows incl. halo
#define LDS_STRIDE 264          // floats/row: [0..2 pad][3 left halo][4..259 main][260 right halo]
#define COL_LEFT   3            // LDS col of left halo  (global x0-1)
#define COL_MAIN   4            // LDS col of main start (global x0), 16B aligned
#define COL_RIGHT  (COL_MAIN + TW)  // 260, right halo (global x0+256)
#define NTHREADS   256

// ---- CDNA5 async global->LDS copies ---------------------------------------
__device__ __forceinline__ void async_copy_b128(uint32_t lds_byte_off,
                                                const float* gptr) {
    asm volatile("global_load_async_to_lds_b128 %0, %1, off"
                 :: "v"(lds_byte_off), "v"(gptr)
                 : "memory");
}

__device__ __forceinline__ void async_copy_b32(uint32_t lds_byte_off,
                                               const float* gptr) {
    asm volatile("global_load_async_to_lds_b32 %0, %1, off"
                 :: "v"(lds_byte_off), "v"(gptr)
                 : "memory");
}

__device__ __forceinline__ void wait_async_zero() {
#if __has_builtin(__builtin_amdgcn_s_wait_asynccnt)
    __builtin_amdgcn_s_wait_asynccnt(0);
#else
    asm volatile("s_wait_asynccnt 0x0" ::: "memory");
#endif
}

// ---- hardware transcendentals (single v_sqrt_f32 / v_rsq_f32, <=2 ULP) ----
__device__ __forceinline__ float fast_sqrt(float x) {
#if __has_builtin(__builtin_amdgcn_sqrtf)
    return __builtin_amdgcn_sqrtf(x);
#else
    return sqrtf(x);
#endif
}

__device__ __forceinline__ float fast_rsqrt(float x) {
#if __has_builtin(__builtin_amdgcn_rsqf)
    return __builtin_amdgcn_rsqf(x);
#else
    return 1.0f / sqrtf(x);
#endif
}

__device__ __forceinline__ int clampi(int v, int lo, int hi) {
    return v < lo ? lo : (v > hi ? hi : v);
}

// ---- main kernel: per-block partial sum of |n_gen - n_tgt| ----------------
__global__ __launch_bounds__(NTHREADS)
void sobel_normals_l1_partial(const float* __restrict__ gen,
                              const float* __restrict__ tgt,
                              float* __restrict__ partials) {
    __shared__ __align__(16) float tile[2][HALO_H][LDS_STRIDE];
    __shared__ float red[NTHREADS];

    const int tid = threadIdx.x;
    const int x0  = blockIdx.x * TW;
    const int y0  = blockIdx.y * TH;
    const size_t img_off = (size_t)blockIdx.z * (size_t)(IMG_HW * IMG_HW);

    const float* src0 = gen + img_off;
    const float* src1 = tgt + img_off;

    // ---- stage halo tiles: interior via b128 (16B/lane), edges via b32 ----
    const int chunk = tid & 63;   // 16B chunk within a row (64 chunks = 256 cols)
    const int rbase = tid >> 6;   // 4 rows per pass
#pragma unroll
    for (int img = 0; img < 2; ++img) {
        const float* src = img ? src1 : src0;
#pragma unroll
        for (int it = 0; it < 5; ++it) {
            const int r = rbase + 4 * it;        // 0..19; rows 18,19 masked off
            if (r < HALO_H) {
                const int gy = clampi(y0 - 1 + r, 0, IMG_HW - 1);
                const float* gp = src + (size_t)gy * IMG_HW + (x0 + 4 * chunk);
                const uint32_t loff =
                    (uint32_t)(uintptr_t)&tile[img][r][COL_MAIN + 4 * chunk];
                async_copy_b128(loff, gp);
            }
        }
    }
    // 2 images x 18 rows x 2 edge columns = 72 scalar copies
    if (tid < 2 * HALO_H * 2) {
        const int side = tid & 1;
        const int r    = (tid >> 1) % HALO_H;
        const int img  = tid / (2 * HALO_H);
        const float* src = img ? src1 : src0;
        const int gy = clampi(y0 - 1 + r, 0, IMG_HW - 1);
        const int gx = side ? clampi(x0 + TW, 0, IMG_HW - 1)
                            : clampi(x0 - 1, 0, IMG_HW - 1);
        const uint32_t loff =
            (uint32_t)(uintptr_t)&tile[img][r][side ? COL_RIGHT : COL_LEFT];
        async_copy_b32(loff, src + (size_t)gy * IMG_HW + gx);
    }
    wait_async_zero();
    __syncthreads();

    // ---- separable Sobel, rolling registers; thread owns column lx --------
    const int lx = tid;  // reads LDS cols lx+3, lx+4, lx+5 (global x-1,x,x+1)
    float dg0, dg1, sg0, sg1, dt0, dt1, st0, st1;
    {
        float a, b, c;
        a = tile[0][0][lx + 3]; b = tile[0][0][lx + 4]; c = tile[0][0][lx + 5];
        dg0 = a - c; sg0 = a + 2.0f * b + c;
        a = tile[0][1][lx + 3]; b = tile[0][1][lx + 4]; c = tile[0][1][lx + 5];
        dg1 = a - c; sg1 = a + 2.0f * b + c;
        a = tile[1][0][lx + 3]; b = tile[1][0][lx + 4]; c = tile[1][0][lx + 5];
        dt0 = a - c; st0 = a + 2.0f * b + c;
        a = tile[1][1][lx + 3]; b = tile[1][1][lx + 4]; c = tile[1][1][lx + 5];
        dt1 = a - c; st1 = a + 2.0f * b + c;
    }

    float acc = 0.0f;
#pragma unroll
    for (int yy = 0; yy < TH; ++yy) {
        const int r = yy + 2;
        float a, b, c;
        a = tile[0][r][lx + 3]; b = tile[0][r][lx + 4]; c = tile[0][r][lx + 5];
        const float dg2 = a - c, sg2 = a + 2.0f * b + c;
        a = tile[1][r][lx + 3]; b = tile[1][r][lx + 4]; c = tile[1][r][lx + 5];
        const float dt2 = a - c, st2 = a + 2.0f * b + c;

        const float gxg = dg0 + 2.0f * dg1 + dg2;   // Sobel-X (cross-corr)
        const float gyg = sg0 - sg2;                // Sobel-Y
        const float gxt = dt0 + 2.0f * dt1 + dt2;
        const float gyt = st0 - st2;

        const float gzg  = 0.25f * fast_sqrt(1.0f - gxg * gxg - gyg * gyg);
        const float invg = fast_rsqrt(4.0f * gxg * gxg + 4.0f * gyg * gyg + gzg * gzg);
        const float gzt  = 0.25f * fast_sqrt(1.0f - gxt * gxt - gyt * gyt);
        const float invt = fast_rsqrt(4.0f * gxt * gxt + 4.0f * gyt * gyt + gzt * gzt);

        acc += fabsf(gxg * invg - gxt * invt)
             + fabsf(gyg * invg - gyt * invt)
             + fabsf(gzg * invg - gzt * invt);

        dg0 = dg1; dg1 = dg2; sg0 = sg1; sg1 = sg2;
        dt0 = dt1; dt1 = dt2; st0 = st1; st1 = st2;
    }

    // ---- deterministic block tree reduction -------------------------------
    red[tid] = acc;
    __syncthreads();
#pragma unroll
    for (int off = NTHREADS / 2; off > 0; off >>= 1) {
        if (tid < off) red[tid] += red[tid + off];
        __syncthreads();
    }
    if (tid == 0) {
        const int bid = ((int)blockIdx.z * (int)gridDim.y + (int)blockIdx.y) *
                            (int)gridDim.x + (int)blockIdx.x;
        partials[bid] = red[0];
    }
}

// ---- final kernel: fixed-order reduction of block partials ----------------
__global__ __launch_bounds__(NTHREADS)
void reduce_partials(const float* __restrict__ partials, int n,
                     float* __restrict__ out, float scale) {
    __shared__ float red[NTHREADS];
    float s = 0.0f;
    for (int i = threadIdx.x; i < n; i += NTHREADS) s += partials[i];
    red[threadIdx.x] = s;
    __syncthreads();
#pragma unroll
    for (int off = NTHREADS / 2; off > 0; off >>= 1) {
        if (threadIdx.x < off) red[threadIdx.x] += red[threadIdx.x + off];
        __syncthreads();
    }
    if (threadIdx.x == 0) out[0] = red[0] * scale;
}

extern "C" void kernel_launch(void* const* d_in, const int* in_sizes, int n_in,
                              void* d_out, int out_size, void* d_ws, size_t ws_size,
                              hipStream_t stream) {
    const float* gen = (const float*)d_in[0];
    const float* tgt = (const float*)d_in[1];
    float* out      = (float*)d_out;
    float* partials = (float*)d_ws;

    const int B = in_sizes[0] / (IMG_HW * IMG_HW);   // 32 for reference shapes
    dim3 grid(IMG_HW / TW, IMG_HW / TH, B);          // (2, 32, B)
    const int nblocks = (int)(grid.x * grid.y * grid.z);

    sobel_normals_l1_partial<<<grid, NTHREADS, 0, stream>>>(gen, tgt, partials);

    const float scale = 1.0f / ((float)B * 3.0f * (float)IMG_HW * (float)IMG_HW);
    reduce_partials<<<1, NTHREADS, 0, stream>>>(partials, nblocks, out, scale);

    (void)n_in; (void)out_size; (void)ws_size;
}